// NodeClassificator_2207613190581
// MI455X (gfx1250) — compile-verified
//
#include <hip/hip_runtime.h>
#include <cmath>

#define HID 128
#define LDSTRIDE 136   // pad so the two half-waves of a B-fragment LDS read hit disjoint banks

typedef __attribute__((ext_vector_type(2))) float v2f;
typedef __attribute__((ext_vector_type(8))) float v8f;

__device__ __forceinline__ float gelu_exact(float x) {
    return 0.5f * x * (1.0f + erff(x * 0.7071067811865476f));
}

__device__ __forceinline__ void atomAddF(float* p, float v) {
    __hip_atomic_fetch_add(p, v, __ATOMIC_RELAXED, __HIP_MEMORY_SCOPE_AGENT);
}

// ---------------- small utility kernels ----------------
__global__ __launch_bounds__(256) void k_fill(float* __restrict__ p, float v, int n) {
    int i = blockIdx.x * 256 + threadIdx.x;
    if (i < n) p[i] = v;
}

__global__ __launch_bounds__(256) void k_fill4(float4* __restrict__ p, int n4) {
    int i = blockIdx.x * 256 + threadIdx.x;
    if (i < n4) p[i] = make_float4(0.f, 0.f, 0.f, 0.f);
}

__global__ __launch_bounds__(256) void k_degree(const int* __restrict__ dst, float* __restrict__ deg, int E) {
    int e = blockIdx.x * 256 + threadIdx.x;
    if (e < E) atomAddF(&deg[dst[e]], 1.0f);
}

__global__ __launch_bounds__(256) void k_rsqrt_inplace(float* __restrict__ d, int n) {
    int i = blockIdx.x * 256 + threadIdx.x;
    if (i < n) d[i] = rsqrtf(fmaxf(d[i], 1.0f));
}

__global__ __launch_bounds__(256) void k_norm(const int* __restrict__ src, const int* __restrict__ dst,
                                              const float* __restrict__ dinv, float* __restrict__ nrm, int E) {
    int e = blockIdx.x * 256 + threadIdx.x;
    if (e < E) nrm[e] = dinv[src[e]] * dinv[dst[e]];
}

// ---------------- sparse propagation: agg += norm[e] * x[src[e]] scattered to dst[e] ----------------
// one wave32 per edge, float4 per lane (128 feats)
__global__ __launch_bounds__(256) void k_spmm(const int* __restrict__ src, const int* __restrict__ dst,
                                              const float* __restrict__ nrm, const float* __restrict__ x,
                                              float* agg, int E) {
    int t = blockIdx.x * 256 + threadIdx.x;
    int e = t >> 5;
    if (e >= E) return;
    int lane = threadIdx.x & 31;
    int s = src[e], d = dst[e];
    float w = nrm[e];
    float4 v = *((const float4*)(x + (size_t)s * HID) + lane);
    float* ap = agg + (size_t)d * HID + lane * 4;
    atomAddF(ap + 0, w * v.x);
    atomAddF(ap + 1, w * v.y);
    atomAddF(ap + 2, w * v.z);
    atomAddF(ap + 3, w * v.w);
}

// ---------------- fused WMMA GEMM (M x 128) @ (128 x 128) ----------------
// MODE 0: out = A @ W + bias                          (lin1)
// MODE 1: A = 0.5*(agg + dinv^2 * x_cur) + 0.5*x0 ;   W' = beta*W + (1-beta)*I ;  out = gelu(A @ W')
//         (out written in place over agg)
// MODE 2: out = gelu(A @ W + bias)                    (fc1)
template<int MODE>
__global__ __launch_bounds__(256) void k_gemm128(const float* A_main, const float* x_cur,
                                                 const float* x0, const float* __restrict__ dinv,
                                                 const float* __restrict__ W, const float* __restrict__ bias,
                                                 float* Out, int M, float beta) {
    __shared__ float sW[HID * LDSTRIDE];
    int tid = threadIdx.x;
    for (int i = tid; i < HID * HID; i += 256) {
        int k = i >> 7, n = i & 127;
        float w = W[i];
        if (MODE == 1) w = beta * w + ((k == n) ? (1.0f - beta) : 0.0f);
        sW[k * LDSTRIDE + n] = w;
    }
    __syncthreads();

    int wave = tid >> 5;
    int lane = tid & 31;
    int rowBase = blockIdx.x * 128 + wave * 16;
    if (rowBase >= M) return;                 // wave-uniform, after barrier: safe

    int hi = lane >> 4;                       // half-wave select (K pair / M half)
    int lm = lane & 15;
    int arow = rowBase + lm;                  // A-fragment row for this lane
    size_t abase = (size_t)arow * HID;

    float selfw = 0.0f;
    if (MODE == 1) { float dv = dinv[arow]; selfw = dv * dv; }

    v8f acc[8];
#pragma unroll
    for (int t = 0; t < 8; ++t) acc[t] = (v8f){0.f, 0.f, 0.f, 0.f, 0.f, 0.f, 0.f, 0.f};

    for (int k0 = 0; k0 < HID; k0 += 4) {
        int ak = k0 + 2 * hi;                 // ISA A layout: lanes0-15 -> {k0,k0+1}, lanes16-31 -> {k0+2,k0+3}
        v2f a;
        if (MODE == 1) {
            float2 ag = *(const float2*)(A_main + abase + ak);
            float2 xc = *(const float2*)(x_cur + abase + ak);
            float2 xz = *(const float2*)(x0 + abase + ak);
            a.x = 0.5f * (ag.x + selfw * xc.x) + 0.5f * xz.x;
            a.y = 0.5f * (ag.y + selfw * xc.y) + 0.5f * xz.y;
        } else {
            float2 av = *(const float2*)(A_main + abase + ak);
            a.x = av.x; a.y = av.y;
        }
#pragma unroll
        for (int nt = 0; nt < 8; ++nt) {
            int n = nt * 16 + lm;
            v2f b;
            b.x = sW[ak * LDSTRIDE + n];
            b.y = sW[(ak + 1) * LDSTRIDE + n];
            acc[nt] = __builtin_amdgcn_wmma_f32_16x16x4_f32(
                false, a, false, b, (short)0, acc[nt], false, false);
        }
    }

    // epilogue: C layout -> m = rowBase + v + 8*hi, n = nt*16 + lm
#pragma unroll
    for (int nt = 0; nt < 8; ++nt) {
        int n = nt * 16 + lm;
#pragma unroll
        for (int v = 0; v < 8; ++v) {
            int m = rowBase + v + 8 * hi;
            float c = acc[nt][v];
            if (MODE == 0)      c = c + bias[n];
            else if (MODE == 1) c = gelu_exact(c);
            else                c = gelu_exact(c + bias[n]);
            Out[(size_t)m * HID + n] = c;
        }
    }
}

// ---------------- layernorm + gelu, one wave32 per row ----------------
__global__ __launch_bounds__(256) void k_ln_gelu(const float* __restrict__ x, const float* __restrict__ g,
                                                 const float* __restrict__ b, float* __restrict__ y, int M) {
    int t = blockIdx.x * 256 + threadIdx.x;
    int row = t >> 5;
    if (row >= M) return;
    int lane = threadIdx.x & 31;
    float4 v = *((const float4*)(x + (size_t)row * HID) + lane);
    float s  = v.x + v.y + v.z + v.w;
    float ss = v.x * v.x + v.y * v.y + v.z * v.z + v.w * v.w;
#pragma unroll
    for (int m = 16; m >= 1; m >>= 1) {
        s  += __shfl_xor(s,  m, 32);
        ss += __shfl_xor(ss, m, 32);
    }
    float mu   = s * (1.0f / 128.0f);
    float var  = ss * (1.0f / 128.0f) - mu * mu;
    float rstd = rsqrtf(var + 1e-5f);
    float4 gv = ((const float4*)g)[lane];
    float4 bv = ((const float4*)b)[lane];
    float4 o;
    o.x = gelu_exact((v.x - mu) * rstd * gv.x + bv.x);
    o.y = gelu_exact((v.y - mu) * rstd * gv.y + bv.y);
    o.z = gelu_exact((v.z - mu) * rstd * gv.z + bv.z);
    o.w = gelu_exact((v.w - mu) * rstd * gv.w + bv.w);
    *((float4*)(y + (size_t)row * HID) + lane) = o;
}

// ---------------- fc2: (M x 128) @ (128 x 10) + b ----------------
__global__ __launch_bounds__(256) void k_fc2(const float* __restrict__ x, const float* __restrict__ w,
                                             const float* __restrict__ b, float* __restrict__ out, int M) {
    int t = blockIdx.x * 256 + threadIdx.x;
    if (t >= M * 10) return;
    int n = t / 10, c = t - n * 10;
    const float* xr = x + (size_t)n * HID;
    float acc = b[c];
#pragma unroll 8
    for (int k = 0; k < HID; ++k) acc = fmaf(xr[k], w[k * 10 + c], acc);
    out[t] = acc;
}

// ---------------- launcher ----------------
extern "C" void kernel_launch(void* const* d_in, const int* in_sizes, int n_in,
                              void* d_out, int out_size, void* d_ws, size_t ws_size,
                              hipStream_t stream) {
    const float* x_in   = (const float*)d_in[0];
    const int*   ei     = (const int*)d_in[1];
    const float* w_lin1 = (const float*)d_in[3];
    const float* b_lin1 = (const float*)d_in[4];
    const float* conv_w = (const float*)d_in[5];
    const float* ln_g   = (const float*)d_in[6];
    const float* ln_b   = (const float*)d_in[7];
    const float* w_fc1  = (const float*)d_in[8];
    const float* b_fc1  = (const float*)d_in[9];
    const float* w_fc2  = (const float*)d_in[10];
    const float* b_fc2  = (const float*)d_in[11];

    const int M = in_sizes[0] / HID;      // 100000
    const int E = in_sizes[1] / 2;        // 1600000
    const int* src = ei;
    const int* dst = ei + E;
    float* out = (float*)d_out;

    // workspace carve-out
    char* ws = (char*)d_ws;
    size_t off = 0;
    auto carve = [&](size_t bytes) -> float* {
        float* p = (float*)(ws + off);
        off = (off + bytes + 255) & ~(size_t)255;
        return p;
    };
    float* dinv  = carve((size_t)M * 4);
    float* nrm   = carve((size_t)E * 4);
    float* x0b   = carve((size_t)M * HID * 4);
    float* bufA  = carve((size_t)M * HID * 4);
    float* bufB  = carve((size_t)M * HID * 4);

    const int TB = 256;
    int gE  = (E + TB - 1) / TB;
    int gM  = (M + TB - 1) / TB;
    int gF4 = (M * (HID / 4) + TB - 1) / TB;       // float4 fill of M*128 floats
    int gSp = (int)(((long long)E * 32 + TB - 1) / TB);
    int gGm = (M + 127) / 128;                     // 128 rows per block
    int gLn = (M * 32 + TB - 1) / TB;
    int gF2 = (M * 10 + TB - 1) / TB;

    // degrees (self loop baked in via init to 1.0), dinv, edge norms
    k_fill<<<gM, TB, 0, stream>>>(dinv, 1.0f, M);
    k_degree<<<gE, TB, 0, stream>>>(dst, dinv, E);
    k_rsqrt_inplace<<<gM, TB, 0, stream>>>(dinv, M);
    k_norm<<<gE, TB, 0, stream>>>(src, dst, dinv, nrm, E);

    // x0 = x @ w_lin1 + b_lin1
    k_gemm128<0><<<gGm, TB, 0, stream>>>(x_in, nullptr, nullptr, nullptr,
                                         w_lin1, b_lin1, x0b, M, 0.0f);

    const float* cur = x0b;
    for (int l = 0; l < 8; ++l) {
        float beta = logf(0.7f / (float)(l + 1) + 1.0f);
        float* nxt = (l & 1) ? bufB : bufA;
        k_fill4<<<gF4, TB, 0, stream>>>((float4*)nxt, M * (HID / 4));
        k_spmm<<<gSp, TB, 0, stream>>>(src, dst, nrm, cur, nxt, E);
        // fused: h = 0.5*(agg + dinv^2*cur) + 0.5*x0 ; out = gelu(h @ (beta*W + (1-beta)I)); in-place over agg
        k_gemm128<1><<<gGm, TB, 0, stream>>>(nxt, cur, x0b, dinv,
                                             conv_w + (size_t)l * HID * HID, nullptr, nxt, M, beta);
        cur = nxt;
    }
    // cur == bufB after layer 7
    float* t1 = (cur == bufA) ? bufB : bufA;
    k_ln_gelu<<<gLn, TB, 0, stream>>>(cur, ln_g, ln_b, t1, M);
    float* t2 = (t1 == bufA) ? bufB : bufA;
    k_gemm128<2><<<gGm, TB, 0, stream>>>(t1, nullptr, nullptr, nullptr,
                                         w_fc1, b_fc1, t2, M, 0.0f);
    k_fc2<<<gF2, TB, 0, stream>>>(t2, w_fc2, b_fc2, out, M);
}